// SPUIGACF_26027501814503
// MI455X (gfx1250) — compile-verified
//
#include <hip/hip_runtime.h>

// Problem constants (match reference)
enum : int { U = 50000, I = 100000, E = 1600000, D = 64, H = 8, HID = 8, B = 16384 };
#define ALPHA 0.2f

typedef float v2f __attribute__((ext_vector_type(2)));
typedef float v8f __attribute__((ext_vector_type(8)));

__device__ __forceinline__ float elu1(float x) { return x > 0.f ? x : __expf(x) - 1.f; }
__device__ __forceinline__ float leaky(float x) { return x >= 0.f ? x : ALPHA * x; }

// Async 16-byte global -> LDS copy (CDNA5 GLOBAL_LOAD_ASYNC_TO_LDS_B128,
// tracked by ASYNCcnt). lds_off is the wave-relative LDS byte address.
__device__ __forceinline__ void async_copy16(unsigned int lds_off, const void* g) {
    asm volatile("global_load_async_to_lds_b128 %0, %1, off"
                 :: "v"(lds_off), "v"((unsigned long long)(uintptr_t)g)
                 : "memory");
}
__device__ __forceinline__ void wait_async0() {
    asm volatile("s_wait_asynccnt 0" ::: "memory");
}

// ---------------------------------------------------------------------------
// W swizzle used by the WMMA GEMM:
//   Ws[(k>>1)*128 + n*2 + (k&1)] = W[k][n]   (k = in-dim 0..63, n = out 0..63)
// so a lane's B fragment (W[k][n], W[k+1][n]) is 8 contiguous bytes in LDS
// -> single ds_load_b64 into an even-aligned VGPR pair, no repacking movs.
// ---------------------------------------------------------------------------

// Fold Wu_h/Wi_h [H,D,HID] -> swizzled [64,64] matrices.
__global__ void k_prepW1(const float* __restrict__ Wu_h, const float* __restrict__ Wi_h,
                         float* __restrict__ W1u, float* __restrict__ W1i) {
    int t = blockIdx.x * blockDim.x + threadIdx.x;   // t = h*512 + d*8 + k
    if (t >= H * D * HID) return;
    int h = t >> 9;
    int d = (t >> 3) & 63;                           // in-dim (GEMM K)
    int k = t & 7;
    int n = h * HID + k;                             // out column
    int dst = (d >> 1) * 128 + n * 2 + (d & 1);
    W1u[dst] = Wu_h[t];
    W1i[dst] = Wi_h[t];
}

// Swizzle row-major [64,64] W_out matrices.
__global__ void k_prepW2(const float* __restrict__ Wu_out, const float* __restrict__ Wi_out,
                         float* __restrict__ W2u, float* __restrict__ W2i) {
    int t = blockIdx.x * blockDim.x + threadIdx.x;   // t = k*64 + n
    if (t >= 64 * 64) return;
    int k = t >> 6, n = t & 63;
    int dst = (k >> 1) * 128 + n * 2 + (k & 1);
    W2u[dst] = Wu_out[t];
    W2i[dst] = Wi_out[t];
}

// ---------------------------------------------------------------------------
// C[nrows,64] = X[nrows,64] @ W[64,64] in fp32 via V_WMMA_F32_16X16X4_F32.
// One wave computes a 16x64 strip: 4 accumulators, 16 K-steps of K=4.
// 256 threads/block = 8 waves = 128 rows. Pre-swizzled W async-staged in LDS.
// WMMA fragment layouts per CDNA5 ISA 7.12.2:
//   A 16x4 f32 : lane L holds row M=(L&15), K = 2*(L>>4) .. +1 in v[0..1]
//   B  4x16 f32: lane L holds col N=(L&15), K = 2*(L>>4) .. +1 in v[0..1]
//   C/D 16x16  : vgpr r, lane L -> row r + 8*(L>>4), col (L&15)
// ---------------------------------------------------------------------------
__global__ void k_gemm64(const float* __restrict__ X, const float* __restrict__ Wswz,
                         float* __restrict__ C, int nrows) {
    __shared__ __align__(16) float Wl[64 * 64];

    // Async-stage 16 KB of pre-swizzled W: 1024 x b128, 16 B per lane.
    for (int t = threadIdx.x; t < 1024; t += 256) {
        unsigned int loff = (unsigned int)(uintptr_t)(&Wl[t * 4]);
        async_copy16(loff, Wswz + t * 4);
    }
    wait_async0();
    __syncthreads();

    const int wave = threadIdx.x >> 5;
    const int lane = threadIdx.x & 31;
    const int m0 = (blockIdx.x * 8 + wave) * 16;
    if (m0 >= nrows) return;                 // uniform per wave: EXEC stays full

    const int rl = lane & 15;
    const int kh = lane >> 4;                // 0 or 1 -> K offset 0 / 2
    const v2f* __restrict__ Xr = (const v2f*)(X + (size_t)(m0 + rl) * 64);
    const v2f* __restrict__ Wf = (const v2f*)Wl;   // 64 float2 per fragment-row

    v8f acc0 = {}, acc1 = {}, acc2 = {}, acc3 = {};
#pragma unroll
    for (int kc = 0; kc < 16; ++kc) {
        v2f a = Xr[kc * 2 + kh];                    // X[row][kb], X[row][kb+1]
        const v2f* wp = Wf + (kc * 2 + kh) * 64;    // fragment-row base
        v2f b0 = wp[rl +  0];
        v2f b1 = wp[rl + 16];
        v2f b2 = wp[rl + 32];
        v2f b3 = wp[rl + 48];
        acc0 = __builtin_amdgcn_wmma_f32_16x16x4_f32(false, a, false, b0, (short)0, acc0, false, false);
        acc1 = __builtin_amdgcn_wmma_f32_16x16x4_f32(false, a, false, b1, (short)0, acc1, false, false);
        acc2 = __builtin_amdgcn_wmma_f32_16x16x4_f32(false, a, false, b2, (short)0, acc2, false, false);
        acc3 = __builtin_amdgcn_wmma_f32_16x16x4_f32(false, a, false, b3, (short)0, acc3, false, false);
    }

    const int r0 = m0 + (kh << 3);
#pragma unroll
    for (int r = 0; r < 8; ++r) {
        float* cr = C + (size_t)(r0 + r) * 64 + rl;
        cr[0]  = acc0[r];
        cr[16] = acc1[r];
        cr[32] = acc2[r];
        cr[48] = acc3[r];
    }
}

// ---------------------------------------------------------------------------
// Layer-1 per-head scores: s[node*8+h] = sum_k xh[node*64+h*8+k] * aH[h*16+aoff+k]
// ---------------------------------------------------------------------------
__global__ void k_scores_head(const float* __restrict__ xh, const float* __restrict__ aH,
                              float* __restrict__ s, int n, int aoff) {
    int t = blockIdx.x * blockDim.x + threadIdx.x;
    if (t >= n * 8) return;
    int node = t >> 3, h = t & 7;
    const float* x = xh + (size_t)node * 64 + h * 8;
    const float* a = aH + h * 16 + aoff;
    float v = 0.f;
#pragma unroll
    for (int k = 0; k < 8; ++k) v += x[k] * a[k];
    s[t] = v;
}

// Layer-2 scalar scores: s[node] = sum_c xh[node*64+c] * av[c]
__global__ void k_scores_out(const float* __restrict__ xh, const float* __restrict__ av,
                             float* __restrict__ s, int n) {
    int node = blockIdx.x * blockDim.x + threadIdx.x;
    if (node >= n) return;
    const float* x = xh + (size_t)node * 64;
    float v = 0.f;
#pragma unroll
    for (int c = 0; c < 64; ++c) v += x[c] * av[c];
    s[node] = v;
}

// ---------------------------------------------------------------------------
// Layer-1 per-(edge,head) row/col sums of e = exp(-leaky(su+si))
// ---------------------------------------------------------------------------
__global__ void k_edge_sums1(const int* __restrict__ eu, const int* __restrict__ ei,
                             const float* __restrict__ su, const float* __restrict__ si,
                             float* __restrict__ row, float* __restrict__ col, int e8) {
    int t = blockIdx.x * blockDim.x + threadIdx.x;
    if (t >= e8) return;
    int e = t >> 3, h = t & 7;
    int u = eu[e], it = ei[e];
    float ev = __expf(-leaky(su[u * 8 + h] + si[it * 8 + h]));
    atomicAdd(&row[u * 8 + h], ev);
    atomicAdd(&col[it * 8 + h], ev);
}

// Layer-1 per-(edge, feature c in 0..63) scatter of e * feature (both directions)
__global__ void k_edge_agg1(const int* __restrict__ eu, const int* __restrict__ ei,
                            const float* __restrict__ su, const float* __restrict__ si,
                            const float* __restrict__ uh, const float* __restrict__ ih,
                            float* __restrict__ aggU, float* __restrict__ aggI, int e64) {
    int t = blockIdx.x * blockDim.x + threadIdx.x;
    if (t >= e64) return;
    int e = t >> 6, c = t & 63, h = c >> 3;
    int u = eu[e], it = ei[e];
    float ev = __expf(-leaky(su[u * 8 + h] + si[it * 8 + h]));
    atomicAdd(&aggU[(size_t)u * 64 + c], ev * ih[(size_t)it * 64 + c]);
    atomicAdd(&aggI[(size_t)it * 64 + c], ev * uh[(size_t)u * 64 + c]);
}

// Layer-2 per-edge sums (single head)
__global__ void k_edge_sums2(const int* __restrict__ eu, const int* __restrict__ ei,
                             const float* __restrict__ su, const float* __restrict__ si,
                             float* __restrict__ row, float* __restrict__ col, int ne) {
    int e = blockIdx.x * blockDim.x + threadIdx.x;
    if (e >= ne) return;
    int u = eu[e], it = ei[e];
    float ev = __expf(-leaky(su[u] + si[it]));
    atomicAdd(&row[u], ev);
    atomicAdd(&col[it], ev);
}

// Layer-2 per-(edge, feature) scatter
__global__ void k_edge_agg2(const int* __restrict__ eu, const int* __restrict__ ei,
                            const float* __restrict__ su, const float* __restrict__ si,
                            const float* __restrict__ uh, const float* __restrict__ ih,
                            float* __restrict__ aggU, float* __restrict__ aggI, int e64) {
    int t = blockIdx.x * blockDim.x + threadIdx.x;
    if (t >= e64) return;
    int e = t >> 6, c = t & 63;
    int u = eu[e], it = ei[e];
    float ev = __expf(-leaky(su[u] + si[it]));
    atomicAdd(&aggU[(size_t)u * 64 + c], ev * ih[(size_t)it * 64 + c]);
    atomicAdd(&aggI[(size_t)it * 64 + c], ev * uh[(size_t)u * 64 + c]);
}

// ---------------------------------------------------------------------------
// Finalize: users get  elu(xh + agg / max(rowsum,~)),
//           items get  elu(xh + (colsum>0 ? agg/colsum : 0)).
// NH = 8 (layer 1 per-head sums) or 1 (layer 2 scalar sums).
// ---------------------------------------------------------------------------
template <int NH>
__global__ void k_final_u(const float* __restrict__ xh, const float* __restrict__ agg,
                          const float* __restrict__ rsum, float* __restrict__ out, int n) {
    int t = blockIdx.x * blockDim.x + threadIdx.x;
    if (t >= n * 64) return;
    int node = t >> 6, c = t & 63;
    float rs = rsum[node * NH + (NH == 8 ? (c >> 3) : 0)];
    float v = xh[t] + agg[t] / (rs == 0.f ? 1.f : rs);
    out[t] = elu1(v);
}

template <int NH>
__global__ void k_final_i(const float* __restrict__ xh, const float* __restrict__ agg,
                          const float* __restrict__ csum, float* __restrict__ out, int n) {
    int t = blockIdx.x * blockDim.x + threadIdx.x;
    if (t >= n * 64) return;
    int node = t >> 6, c = t & 63;
    float cs = csum[node * NH + (NH == 8 ? (c >> 3) : 0)];
    float att = cs > 0.f ? agg[t] / cs : 0.f;
    out[t] = elu1(xh[t] + att);
}

// ---------------------------------------------------------------------------
// out[b] = dot64(u_f[userIdx[b]], i_f[itemIdx[b]]) — one wave32 per output.
// ---------------------------------------------------------------------------
__global__ void k_dot(const float* __restrict__ uf, const float* __restrict__ itf,
                      const int* __restrict__ uidx, const int* __restrict__ iidx,
                      float* __restrict__ out, int nb) {
    int gtid = blockIdx.x * blockDim.x + threadIdx.x;
    int wid = gtid >> 5;
    int lane = threadIdx.x & 31;
    if (wid >= nb) return;
    const float* a = uf + (size_t)uidx[wid] * 64;
    const float* b = itf + (size_t)iidx[wid] * 64;
    float v = a[lane] * b[lane] + a[lane + 32] * b[lane + 32];
#pragma unroll
    for (int off = 16; off > 0; off >>= 1) v += __shfl_xor(v, off, 32);
    if (lane == 0) out[wid] = v;
}

// ---------------------------------------------------------------------------
extern "C" void kernel_launch(void* const* d_in, const int* in_sizes, int n_in,
                              void* d_out, int out_size, void* d_ws, size_t ws_size,
                              hipStream_t stream) {
    const int*   userIdx = (const int*)d_in[0];
    const int*   itemIdx = (const int*)d_in[1];
    const int*   edge_u  = (const int*)d_in[2];
    const int*   edge_i  = (const int*)d_in[3];
    const float* uEmbd   = (const float*)d_in[4];
    const float* iEmbd   = (const float*)d_in[5];
    const float* Wu_h    = (const float*)d_in[6];
    const float* Wi_h    = (const float*)d_in[7];
    const float* a_h     = (const float*)d_in[8];
    const float* Wu_out  = (const float*)d_in[9];
    const float* Wi_out  = (const float*)d_in[10];
    const float* a_out   = (const float*)d_in[11];
    float* out = (float*)d_out;

    // ---- workspace carve-up (floats); total ~31.2M floats ~125 MB ----
    float* ws = (float*)d_ws;
    size_t o = 0;
    float* W1u  = ws + o; o += 4096;             // swizzled layer-1 W (users)
    float* W1i  = ws + o; o += 4096;
    float* W2u  = ws + o; o += 4096;             // swizzled layer-2 W
    float* W2i  = ws + o; o += 4096;
    float* u_h  = ws + o; o += (size_t)U * 64;   // layer1 then layer2 features
    float* i_h  = ws + o; o += (size_t)I * 64;
    float* su   = ws + o; o += (size_t)U * 8;    // layer1 per-head / layer2 scalar
    float* si   = ws + o; o += (size_t)I * 8;
    float* row  = ws + o; o += (size_t)U * 8;    // zeroed region starts here
    float* col  = ws + o; o += (size_t)I * 8;
    float* aggU = ws + o; o += (size_t)U * 64;
    float* aggI = ws + o; o += (size_t)I * 64;   // zeroed region ends here
    float* hu   = ws + o; o += (size_t)U * 64;   // layer1 hidden -> reused as u_f
    float* hi   = ws + o; o += (size_t)I * 64;   // layer1 hidden -> reused as i_f
    (void)ws_size; (void)in_sizes; (void)n_in; (void)out_size;

    const size_t zero_bytes =
        ((size_t)U * 8 + (size_t)I * 8 + (size_t)U * 64 + (size_t)I * 64) * sizeof(float);

    const int TB = 256;
    const dim3 blk(TB);

    // ===== Layer 1 =====
    k_prepW1<<<dim3((H * D * HID + TB - 1) / TB), blk, 0, stream>>>(Wu_h, Wi_h, W1u, W1i);
    k_prepW2<<<dim3((64 * 64 + TB - 1) / TB), blk, 0, stream>>>(Wu_out, Wi_out, W2u, W2i);
    hipMemsetAsync(row, 0, zero_bytes, stream);

    k_gemm64<<<dim3((U + 127) / 128), blk, 0, stream>>>(uEmbd, W1u, u_h, U);
    k_gemm64<<<dim3((I + 127) / 128), blk, 0, stream>>>(iEmbd, W1i, i_h, I);

    k_scores_head<<<dim3((U * 8 + TB - 1) / TB), blk, 0, stream>>>(u_h, a_h, su, U, 0);
    k_scores_head<<<dim3((I * 8 + TB - 1) / TB), blk, 0, stream>>>(i_h, a_h, si, I, HID);

    k_edge_sums1<<<dim3((E * 8 + TB - 1) / TB), blk, 0, stream>>>(edge_u, edge_i, su, si, row, col, E * 8);
    k_edge_agg1<<<dim3((E * 64 + TB - 1) / TB), blk, 0, stream>>>(edge_u, edge_i, su, si, u_h, i_h, aggU, aggI, E * 64);

    k_final_u<8><<<dim3((U * 64 + TB - 1) / TB), blk, 0, stream>>>(u_h, aggU, row, hu, U);
    k_final_i<8><<<dim3((I * 64 + TB - 1) / TB), blk, 0, stream>>>(i_h, aggI, col, hi, I);

    // ===== Layer 2 =====
    hipMemsetAsync(row, 0, zero_bytes, stream);

    k_gemm64<<<dim3((U + 127) / 128), blk, 0, stream>>>(hu, W2u, u_h, U);
    k_gemm64<<<dim3((I + 127) / 128), blk, 0, stream>>>(hi, W2i, i_h, I);

    k_scores_out<<<dim3((U + TB - 1) / TB), blk, 0, stream>>>(u_h, a_out, su, U);
    k_scores_out<<<dim3((I + TB - 1) / TB), blk, 0, stream>>>(i_h, a_out + D, si, I);

    k_edge_sums2<<<dim3((E + TB - 1) / TB), blk, 0, stream>>>(edge_u, edge_i, su, si, row, col, E);
    k_edge_agg2<<<dim3((E * 64 + TB - 1) / TB), blk, 0, stream>>>(edge_u, edge_i, su, si, u_h, i_h, aggU, aggI, E * 64);

    // final ELU; write u_f/i_f over hu/hi (no longer needed)
    k_final_u<1><<<dim3((U * 64 + TB - 1) / TB), blk, 0, stream>>>(u_h, aggU, row, hu, U);
    k_final_i<1><<<dim3((I * 64 + TB - 1) / TB), blk, 0, stream>>>(i_h, aggI, col, hi, I);

    // ===== Output dot products =====
    k_dot<<<dim3((B * 32 + TB - 1) / TB), blk, 0, stream>>>(hu, hi, userIdx, itemIdx, out, B);
}